// Motion_loss_66657892434040
// MI455X (gfx1250) — compile-verified
//
#include <hip/hip_runtime.h>
#include <stdint.h>

typedef unsigned int v4u __attribute__((ext_vector_type(4)));
typedef int          v8i __attribute__((ext_vector_type(8)));
typedef int          v4i __attribute__((ext_vector_type(4)));

#define NB      64      // batch
#define NT      2048    // time steps
#define NJ      24      // joints
#define NC      99      // channels = 4*NJ + 3
#define TILE_T  256     // t-samples per block (one per thread)
#define CHUNK_R 12      // rows per TDM chunk (3 joints * 4 quat rows)
#define NCHUNK  9       // 8 quat chunks (96 rows) + 1 position chunk (3 rows)
#define TILES_PER_B (NT / TILE_T)   // 8

static constexpr int TOPO[NJ] =
  {-1,0,0,0,1,2,3,4,5,6,7,8,9,9,9,12,13,14,16,17,18,19,20,21};

__device__ static inline void quat_to_mat(const float q[4], float m[9]) {
  const float w = q[0], x = q[1], y = q[2], z = q[3];
  const float s = w*w + x*x + y*y + z*z;
  const float ts = 2.0f / s;
  m[0] = 1.0f - ts*(y*y + z*z);
  m[1] = ts*(x*y - z*w);
  m[2] = ts*(x*z + y*w);
  m[3] = ts*(x*y + z*w);
  m[4] = 1.0f - ts*(x*x + z*z);
  m[5] = ts*(y*z - x*w);
  m[6] = ts*(x*z - y*w);
  m[7] = ts*(y*z + x*w);
  m[8] = 1.0f - ts*(x*x + y*y);
}

__device__ static inline void mat33_mul(const float a[9], const float b[9], float c[9]) {
#pragma unroll
  for (int i = 0; i < 3; ++i)
#pragma unroll
    for (int j = 0; j < 3; ++j)
      c[3*i+j] = a[3*i+0]*b[0+j] + a[3*i+1]*b[3+j] + a[3*i+2]*b[6+j];
}

__device__ static inline void mat33_vec(const float a[9], const float v[3], float r[3]) {
  r[0] = a[0]*v[0] + a[1]*v[1] + a[2]*v[2];
  r[1] = a[3]*v[0] + a[4]*v[1] + a[5]*v[2];
  r[2] = a[6]*v[0] + a[7]*v[1] + a[8]*v[2];
}

// Issue one TDM 2D tile load: rows x TILE_T floats, row pitch NT floats,
// from raw[b, c0:c0+rows, t0:t0+TILE_T] into LDS at byte offset lds_byte.
// Descriptor layout per CDNA5 ISA ch.8 (D# groups 0/1; groups 2/3 zero => 2D).
__device__ static inline void tdm_load_chunk(const float* gbase, unsigned lds_byte,
                                             int b, int c0, int t0, int rows) {
  const uint64_t ga = (uint64_t)(uintptr_t)gbase +
      4ull * ((uint64_t)b * NC * NT + (uint64_t)c0 * NT + (uint64_t)t0);
  v4u g0;
  g0.x = 1u;                                                  // count=1 (valid), user mode
  g0.y = lds_byte;                                            // lds_addr (bytes)
  g0.z = (unsigned)(ga & 0xFFFFFFFFull);                      // global_addr[31:0]
  g0.w = (unsigned)((ga >> 32) & 0x1FFFFFFull) | (2u << 30);  // addr[56:32] | type=2

  const unsigned d0 = (unsigned)(NT - t0);   // tensor extent x (OOB bound)
  const unsigned d1 = (unsigned)(NC - c0);   // tensor extent y
  v8i g1;
  g1[0] = (int)(2u << 16);                                   // wg_mask=0, data_size=4B
  g1[1] = (int)((d0 & 0xFFFFu) << 16);                       // barrier_addr=0 | dim0.lo
  g1[2] = (int)((d0 >> 16) | ((d1 & 0xFFFFu) << 16));        // dim0.hi | dim1.lo
  g1[3] = (int)((d1 >> 16) | ((unsigned)TILE_T << 16));      // dim1.hi | tile_dim0=256
  g1[4] = rows;                                              // tile_dim1=rows, tile_dim2=0
  g1[5] = (int)NT;                                           // tensor_dim0_stride.lo (elems)
  g1[6] = 0;                                                 // stride.hi | dim1_stride.lo
  g1[7] = 0;
  v4i zero4 = {0, 0, 0, 0};
  v8i zero8 = {0, 0, 0, 0, 0, 0, 0, 0};
  // 6-arg builtin on this toolchain: (g0, g1, g2, g3, ext, cpol)
  __builtin_amdgcn_tensor_load_to_lds(g0, g1, zero4, zero4, zero8, 0);
}

__global__ __launch_bounds__(TILE_T) void motion_loss_main(
    const float* __restrict__ Ym, const float* __restrict__ Xm,
    const float* __restrict__ Yt, const float* __restrict__ Xt,
    const float* __restrict__ W, float* __restrict__ part)
{
  __shared__ float s_tiles[2][2][CHUNK_R * TILE_T];  // [buf][Y/X][row*TILE_T] = 48 KB
  __shared__ float s_offY[NJ * 3];
  __shared__ float s_offX[NJ * 3];
  __shared__ float s_w[NJ];
  __shared__ float s_red[3];

  const int tid = threadIdx.x;
  const int b   = (int)blockIdx.x / TILES_PER_B;
  const int t0  = ((int)blockIdx.x % TILES_PER_B) * TILE_T;

  if (tid < NJ * 3) { s_offY[tid] = Yt[b * NJ * 3 + tid]; s_offX[tid] = Xt[b * NJ * 3 + tid]; }
  if (tid < NJ)     s_w[tid]   = W[tid];
  if (tid < 3)      s_red[tid] = 0.0f;

  const bool wave0 = (tid < 32);
  const unsigned chunkBytes = CHUNK_R * TILE_T * 4u;

  if (wave0) {   // prologue: chunk 0 -> buffer 0 (TDM ignores EXEC; per-wave TENSORcnt)
    const unsigned base = (unsigned)(uintptr_t)(const void*)&s_tiles[0][0][0];
    tdm_load_chunk(Ym, base,              b, 0, t0, CHUNK_R);
    tdm_load_chunk(Xm, base + chunkBytes, b, 0, t0, CHUNK_R);
  }

  float accR = 0.0f, accF = 0.0f, accP = 0.0f;
  float Ty[NJ][9], Tx[NJ][9];     // chained transforms (SROA'd after full unroll)
  float ry[NJ][3], rx[NJ][3];     // world positions

#pragma unroll
  for (int ck = 0; ck < NCHUNK; ++ck) {
    if (wave0) {
      if (ck + 1 < NCHUNK) {
        const int c0n   = (ck + 1) * CHUNK_R;
        const int rowsn = (ck + 1 == NCHUNK - 1) ? (NC - c0n) : CHUNK_R;
        const unsigned base =
            (unsigned)(uintptr_t)(const void*)&s_tiles[(ck + 1) & 1][0][0];
        tdm_load_chunk(Ym, base,              b, c0n, t0, rowsn);
        tdm_load_chunk(Xm, base + chunkBytes, b, c0n, t0, rowsn);
        __builtin_amdgcn_s_wait_tensorcnt(2);   // current chunk done; next in flight
      } else {
        __builtin_amdgcn_s_wait_tensorcnt(0);
      }
    }
    __syncthreads();

    const float* bufY = &s_tiles[ck & 1][0][0];
    const float* bufX = &s_tiles[ck & 1][1][0];

    if (ck < 8) {
#pragma unroll
      for (int jj = 0; jj < 3; ++jj) {
        const int j  = ck * 3 + jj;
        const int pi = TOPO[j];
        float qy[4], qx[4];
#pragma unroll
        for (int k = 0; k < 4; ++k) {           // lanes hit consecutive LDS banks
          qy[k] = bufY[(4 * jj + k) * TILE_T + tid];
          qx[k] = bufX[(4 * jj + k) * TILE_T + tid];
        }
        const float wj = s_w[j];
        accR += wj * (fabsf(qy[0]-qx[0]) + fabsf(qy[1]-qx[1]) +
                      fabsf(qy[2]-qx[2]) + fabsf(qy[3]-qx[3]));

        float Ry[9], Rx[9];
        quat_to_mat(qy, Ry);
        quat_to_mat(qx, Rx);
        if (pi < 0) {
#pragma unroll
          for (int k = 0; k < 9; ++k) { Ty[j][k] = Ry[k]; Tx[j][k] = Rx[k]; }
        } else {
          mat33_mul(Ty[pi], Ry, Ty[j]);
          mat33_mul(Tx[pi], Rx, Tx[j]);
        }
        const float oy[3] = { s_offY[3*j], s_offY[3*j+1], s_offY[3*j+2] };
        const float ox[3] = { s_offX[3*j], s_offX[3*j+1], s_offX[3*j+2] };
        mat33_vec(Ty[j], oy, ry[j]);
        mat33_vec(Tx[j], ox, rx[j]);
        if (pi >= 0) {
#pragma unroll
          for (int k = 0; k < 3; ++k) { ry[j][k] += ry[pi][k]; rx[j][k] += rx[pi][k]; }
        }
        accF += wj * (fabsf(ry[j][0]-rx[j][0]) + fabsf(ry[j][1]-rx[j][1]) +
                      fabsf(ry[j][2]-rx[j][2]));
      }
    } else {    // position rows (c = 96..98), unweighted
#pragma unroll
      for (int k = 0; k < 3; ++k)
        accP += fabsf(bufX[k * TILE_T + tid] - bufY[k * TILE_T + tid]);
    }
    __syncthreads();   // buffer (ck&1) free for reuse before next prefetch lands
  }

  // fold the means in per-thread (keeps the float sums small)
  accR *= (1.0f / ((float)NB * NT * NJ * 4));
  accF *= (1.0f / ((float)NB * NT * NJ * 3));
  accP *= (1.0f / ((float)NB * NT * 3));

  // wave32 reduction tree
#pragma unroll
  for (int off = 16; off > 0; off >>= 1) {
    accR += __shfl_down(accR, off, 32);
    accF += __shfl_down(accF, off, 32);
    accP += __shfl_down(accP, off, 32);
  }
  if ((tid & 31) == 0) {          // one ds_add_f32 per wave
    atomicAdd(&s_red[0], accR);
    atomicAdd(&s_red[1], accF);
    atomicAdd(&s_red[2], accP);
  }
  __syncthreads();
  if (tid == 0) {                 // one global_atomic_add_f32 per block
    atomicAdd(&part[0], s_red[0]);
    atomicAdd(&part[1], s_red[1]);
    atomicAdd(&part[2], s_red[2]);
  }
}

__global__ void motion_loss_finalize(const float* __restrict__ part,
                                     float* __restrict__ out) {
  if (threadIdx.x == 0 && blockIdx.x == 0) {
    const float B1 = 1.0f, B2 = 1.0f;
    out[0] = B1 * part[0] + B2 * part[1] + part[2];
  }
}

extern "C" void kernel_launch(void* const* d_in, const int* in_sizes, int n_in,
                              void* d_out, int out_size, void* d_ws, size_t ws_size,
                              hipStream_t stream) {
  (void)in_sizes; (void)n_in; (void)out_size; (void)ws_size;
  const float* Ym = (const float*)d_in[0];   // (B, C, T) f32
  const float* Xm = (const float*)d_in[1];   // (B, C, T) f32
  const float* Yt = (const float*)d_in[2];   // (B, J, 3) f32
  const float* Xt = (const float*)d_in[3];   // (B, J, 3) f32
  const float* W  = (const float*)d_in[4];   // (J, 1)    f32
  float* part = (float*)d_ws;                // 3 partial sums
  float* out  = (float*)d_out;

  (void)hipMemsetAsync(part, 0, 3 * sizeof(float), stream);  // capture-safe memset node
  dim3 grid(NB * TILES_PER_B);                               // 512 blocks
  dim3 block(TILE_T);                                        // 256 threads = 8 wave32
  motion_loss_main<<<grid, block, 0, stream>>>(Ym, Xm, Yt, Xt, W, part);
  motion_loss_finalize<<<1, 32, 0, stream>>>(part, out);
}